// FeatureTransformer_49374944035275
// MI455X (gfx1250) — compile-verified
//
#include <hip/hip_runtime.h>

typedef __attribute__((ext_vector_type(2))) float v2f;
typedef __attribute__((ext_vector_type(8))) float v8f;

#define HID 256          // HIDDEN_SIZE (reference constant)
#define BAGS_PER_WG 16   // one 16-row WMMA M-tile of bags per workgroup
#define ROW_BLK 512      // gathered rows staged in LDS per block

// D = A(16x4 one-hot selection) x B(4 gathered rows x 16 cols) + C, fp32 WMMA.
__global__ __launch_bounds__(256, 2)
void embag_wmma_kernel(const int* __restrict__ fidx,
                       const int* __restrict__ offsets,
                       const float* __restrict__ table,
                       const float* __restrict__ bias,
                       float* __restrict__ out,
                       int n_total, int batch)
{
    __shared__ int s_off[BAGS_PER_WG + 1];
    __shared__ int s_fi[ROW_BLK];
    __shared__ int s_sg[ROW_BLK];

    const int tid      = threadIdx.x;
    const int bag_base = blockIdx.x * BAGS_PER_WG;

    // Stage the 17 bag boundaries for this tile (clamped to n_total).
    if (tid <= BAGS_PER_WG) {
        int b = bag_base + tid;
        s_off[tid] = (b < batch) ? offsets[b] : n_total;
    }
    __syncthreads();

    const int row_begin = s_off[0];
    const int row_end   = s_off[BAGS_PER_WG];

    const int lane  = tid & 31;
    const int wave  = tid >> 5;
    const int mrow  = lane & 15;   // M for A-matrix, N for B/C/D matrices
    const int nhalf = lane >> 4;   // lane half: selects K pair / M half
    const int k0sel = nhalf * 2;   // A/B VGPR0 holds K=0 (lanes 0-15) / K=2 (lanes 16-31)
    const int col0  = wave * 32 + mrow;  // first 16-wide hidden tile for this wave
    const int col1  = col0 + 16;         // second 16-wide hidden tile

    // C init = bias[n]; bias depends only on N, identical across the 8 C VGPRs.
    v8f acc0, acc1;
    {
        float b0 = bias[col0];
        float b1 = bias[col1];
#pragma unroll
        for (int j = 0; j < 8; ++j) { acc0[j] = b0; acc1[j] = b1; }
    }

    for (int blk = row_begin; blk < row_end; blk += ROW_BLK) {
        const int blk_rows = min(ROW_BLK, row_end - blk);
        const int padded   = (blk_rows + 3) & ~3;

        // Cooperatively stage row indices + per-row segment ids in LDS.
        for (int i = tid; i < padded; i += 256) {
            int fi = 0, sg = -1;          // padding: seg=-1 -> zero A column
            if (i < blk_rows) {
                const int r = blk + i;
                fi = fidx[r];
                sg = 0;
#pragma unroll
                for (int t = 1; t < BAGS_PER_WG; ++t)
                    sg += (r >= s_off[t]) ? 1 : 0;
            }
            s_fi[i] = fi;
            s_sg[i] = sg;
        }
        __syncthreads();

        // WMMA loop: uniform bounds, EXEC all-1s.
        for (int c = 0; c < padded; c += 4) {
            const int fiA = s_fi[c + k0sel];
            const int fiB = s_fi[c + k0sel + 1];
            const int sgA = s_sg[c + k0sel];
            const int sgB = s_sg[c + k0sel + 1];

            // A[m][k] = 1 iff gathered row k belongs to bag m of this tile.
            v2f a;
            a[0] = (sgA == mrow) ? 1.0f : 0.0f;
            a[1] = (sgB == mrow) ? 1.0f : 0.0f;

            // B[k][n]: 4 gathered table rows, 16-column slices per tile.
            const float* rowA = table + (size_t)fiA * HID;
            const float* rowB = table + (size_t)fiB * HID;
            v2f bt0, bt1;
            bt0[0] = rowA[col0]; bt0[1] = rowB[col0];
            bt1[0] = rowA[col1]; bt1[1] = rowB[col1];

            acc0 = __builtin_amdgcn_wmma_f32_16x16x4_f32(
                false, a, false, bt0, (short)0, acc0, false, false);
            acc1 = __builtin_amdgcn_wmma_f32_16x16x4_f32(
                false, a, false, bt1, (short)0, acc1, false, false);
        }
        __syncthreads();
    }

    // D layout: VGPR j -> bag row j (lanes 0-15) / j+8 (lanes 16-31), col = N.
    float* outp = out + (size_t)bag_base * HID;
    const int mlim = batch - bag_base;
#pragma unroll
    for (int j = 0; j < 8; ++j) {
        int m = j + nhalf * 8;
        if (m < mlim) {
            outp[(size_t)m * HID + col0] = acc0[j];
            outp[(size_t)m * HID + col1] = acc1[j];
        }
    }
}

extern "C" void kernel_launch(void* const* d_in, const int* in_sizes, int n_in,
                              void* d_out, int out_size, void* d_ws, size_t ws_size,
                              hipStream_t stream) {
    (void)n_in; (void)out_size; (void)d_ws; (void)ws_size;
    const int*   fidx  = (const int*)d_in[0];
    const int*   offs  = (const int*)d_in[1];
    const float* table = (const float*)d_in[2];
    const float* bias  = (const float*)d_in[3];
    float*       out   = (float*)d_out;

    const int n_total = in_sizes[0];   // BATCH * BAG flat indices
    const int batch   = in_sizes[1];   // number of bags

    const int grid = (batch + BAGS_PER_WG - 1) / BAGS_PER_WG;
    embag_wmma_kernel<<<grid, 256, 0, stream>>>(fidx, offs, table, bias, out,
                                                n_total, batch);
}